// Muti_DiceLoss_79456894976115
// MI455X (gfx1250) — compile-verified
//
#include <hip/hip_runtime.h>
#include <hip/hip_bf16.h>
#include <math.h>

namespace {

constexpr int kB      = 8;
constexpr int kC      = 12;
constexpr int kMaps   = 6;
constexpr int kHW     = 640 * 640;       // 409600
constexpr int kHW4    = kHW / 4;         // 102400 float4 per channel
constexpr int kPairs  = 16;              // B * N_CLASSES
constexpr int kNB     = 100;             // streaming blocks per pair
constexpr int kT      = 256;             // threads per block (8 waves, wave32)
constexpr int kF4PerBlock  = kHW4 / kNB;        // 1024 float4 = 16KB per stream
constexpr int kF4PerThread = kF4PerBlock / kT;  // 4
constexpr float kLambda = 0.7f;
constexpr float kEps    = 1e-4f;
constexpr int  kRatio   = 3;

constexpr int BINS_A = 2048;  // stages 0,1 (11 bits each)
constexpr int BINS_B = 1024;  // stage 2 (10 bits)

// ---- workspace layout (32-bit word offsets into d_ws) ----
constexpr int POS_OFF   = 0;
constexpr int NEG_OFF   = POS_OFF + kPairs;
constexpr int HIST_OFF  = NEG_OFF + kPairs;               // kPairs * BINS_A u32
constexpr int FB_OFF    = HIST_OFF + kPairs * BINS_A;
constexpr int RANK_OFF  = FB_OFF   + kPairs;
constexpr int PREF_OFF  = RANK_OFF + kPairs;
constexpr int THR_OFF   = PREF_OFF + kPairs;              // f32 view
constexpr int KPART_OFF = THR_OFF  + kPairs;              // f32: kPairs*kNB*15
constexpr int TPART_OFF = KPART_OFF + kPairs * kNB * 15;  // f32: kPairs*kNB*3
constexpr int ZERO_WORDS = FB_OFF;                        // counts + hist must be zeroed

__device__ __forceinline__ float fget(const float4& v, int j) {
  return j == 0 ? v.x : j == 1 ? v.y : j == 2 ? v.z : v.w;
}
__device__ __forceinline__ float sigm(float x) { return 1.0f / (1.0f + expf(-x)); }

// monotone float -> u32 key (ascending key order == ascending float order)
__device__ __forceinline__ unsigned keymap(float f) {
  unsigned u = __float_as_uint(f);
  return (u & 0x80000000u) ? ~u : (u | 0x80000000u);
}
__device__ __forceinline__ float keyunmap(unsigned k) {
  unsigned u = (k & 0x80000000u) ? (k & 0x7FFFFFFFu) : ~k;
  return __uint_as_float(u);
}

// fixed-order block tree reduction (deterministic across replays)
__device__ __forceinline__ float blockReduceF(float v, float* buf) {
  int tid = threadIdx.x;
  buf[tid] = v; __syncthreads();
  for (int s = kT / 2; s > 0; s >>= 1) {
    if (tid < s) buf[tid] += buf[tid + s];
    __syncthreads();
  }
  float r = buf[0]; __syncthreads();
  return r;
}
__device__ __forceinline__ unsigned blockReduceU(unsigned v, unsigned* buf) {
  int tid = threadIdx.x;
  buf[tid] = v; __syncthreads();
  for (int s = kT / 2; s > 0; s >>= 1) {
    if (tid < s) buf[tid] += buf[tid + s];
    __syncthreads();
  }
  unsigned r = buf[0]; __syncthreads();
  return r;
}

// ---------------------------------------------------------------------------
// Tensor Data Mover (gfx1250): async DMA of a 1-D f32 tile global -> LDS.
// Arity differs: AMD clang-22 (5 args) vs upstream clang-23 (6 args).
#if __has_builtin(__builtin_amdgcn_tensor_load_to_lds) && \
    __has_builtin(__builtin_amdgcn_s_wait_tensorcnt)
#define HAVE_TDM 1
typedef __attribute__((ext_vector_type(4))) unsigned tdm_v4u;
typedef __attribute__((ext_vector_type(8))) int      tdm_v8i;
typedef __attribute__((ext_vector_type(4))) int      tdm_v4i;

__device__ __forceinline__ void tdm_load_f32_1d(const float* gsrc,
                                                unsigned lds_off, unsigned n) {
  const unsigned long long ga = (unsigned long long)(uintptr_t)gsrc;
  tdm_v4u g0;
  g0[0] = 0x1u;                                  // count=1 (valid user D#)
  g0[1] = lds_off;                               // lds_addr (bytes)
  g0[2] = (unsigned)ga;                          // global_addr[31:0]
  g0[3] = (unsigned)((ga >> 32) & 0x01FFFFFFu)   // global_addr[56:32]
          | 0x80000000u;                         // type=2 ("image")
  tdm_v8i g1;
  g1[0] = (int)(2u << 16);                       // data_size=4B; no mcast/pad
  g1[1] = (int)((n & 0xFFFFu) << 16);            // tensor_dim0[15:0]
  g1[2] = (int)(((n >> 16) & 0xFFFFu)            // tensor_dim0[31:16]
          | (1u << 16));                         // tensor_dim1 = 1
  g1[3] = (int)((n & 0xFFFFu) << 16);            // tile_dim0 = n (1-D tile)
  g1[4] = 0;                                     // tile_dim1/2 unused
  g1[5] = (int)n;                                // tensor_dim0_stride[31:0]
  g1[6] = 0;
  g1[7] = 0;
  tdm_v4i z4 = {0, 0, 0, 0};
#if defined(__clang_major__) && (__clang_major__ >= 23)
  tdm_v8i z8 = {0, 0, 0, 0, 0, 0, 0, 0};
  __builtin_amdgcn_tensor_load_to_lds(g0, g1, z4, z4, z8, 0);
#else
  __builtin_amdgcn_tensor_load_to_lds(g0, g1, z4, z4, 0);
#endif
}
__device__ __forceinline__ unsigned lds_offset_of(const void* p) {
  return (unsigned)(uintptr_t)p;                 // flat LDS addr: low 32 bits
}
#else
#define HAVE_TDM 0
#endif

// ---------------------------------------------------------------------------
// Full-precision f32 WMMA (V_WMMA_F32_16X16X4_F32) for the final partial-sum
// reduction: S = R x ones is a real matrix-vector product, and the f32 WMMA
// accumulator keeps it exact-in-f32 and bit-deterministic.
#if __has_builtin(__builtin_amdgcn_wmma_f32_16x16x4_f32)
#define HAVE_WMMA_F32 1
typedef __attribute__((ext_vector_type(2))) float wm_v2f;
typedef __attribute__((ext_vector_type(8))) float wm_v8f;
#else
#define HAVE_WMMA_F32 0
#endif

// ---------------------------------------------------------------------------
__global__ __launch_bounds__(kT) void k_zero(unsigned* __restrict__ ws, int n) {
  int i = blockIdx.x * blockDim.x + threadIdx.x;
  if (i < n) ws[i] = 0u;
}

// ---------------------------------------------------------------------------
// Pass 1: pos/neg counts, level-0 histogram of negative-score keys (top 11
// bits), and all 5*3 kernel-dice partial sums for this block's pixel chunk.
__global__ __launch_bounds__(kT) void k_stats(const float* __restrict__ outp,
                                              const float* __restrict__ labp,
                                              const float* __restrict__ tmp,
                                              unsigned* __restrict__ ws_u,
                                              float* __restrict__ kpart) {
  __shared__ unsigned hist[BINS_A];
  __shared__ float    fbuf[kT];
  __shared__ unsigned ubuf[kT];

  const int pair = blockIdx.y;           // b*2 + c
  const int b = pair >> 1, c = pair & 1;

  for (int i = threadIdx.x; i < BINS_A; i += kT) hist[i] = 0u;
  __syncthreads();

  const int txtCh = c * kMaps + (kMaps - 1);
  const float4* texts = (const float4*)(outp + (size_t)(b * kC + txtCh) * kHW);
  const float4* gtex  = (const float4*)(labp + (size_t)(b * kC + txtCh) * kHW);
  const float4* tmv   = (const float4*)(tmp  + (size_t)b * kHW);
  const float4* kernp[5];
  const float4* kgp[5];
#pragma unroll
  for (int k = 0; k < 5; ++k) {
    kernp[k] = (const float4*)(outp + (size_t)(b * kC + c * kMaps + k) * kHW);
    kgp[k]   = (const float4*)(labp + (size_t)(b * kC + c * kMaps + k) * kHW);
  }

  const int base = blockIdx.x * kF4PerBlock;
  unsigned posc = 0, negc = 0;
  float ks[15];
#pragma unroll
  for (int v = 0; v < 15; ++v) ks[v] = 0.0f;

  for (int it = 0; it < kF4PerThread; ++it) {
    const int idx = base + it * kT + threadIdx.x;
    if (it + 1 < kF4PerThread) {           // gfx1250 global_prefetch_b8 path
      __builtin_prefetch(&texts[idx + kT], 0, 0);
      __builtin_prefetch(&gtex[idx + kT], 0, 0);
      __builtin_prefetch(&tmv[idx + kT], 0, 0);
    }
    const float4 s4 = texts[idx];
    const float4 g4 = gtex[idx];
    const float4 m4 = tmv[idx];
    float4 kp4[5], kg4[5];
#pragma unroll
    for (int k = 0; k < 5; ++k) { kp4[k] = kernp[k][idx]; kg4[k] = kgp[k][idx]; }

#pragma unroll
    for (int j = 0; j < 4; ++j) {
      const float s = fget(s4, j);
      const float g = fget(g4, j);
      const float m = fget(m4, j);
      posc += (g > 0.5f && m > 0.5f) ? 1u : 0u;
      if (g <= 0.5f) {
        ++negc;
        atomicAdd(&hist[keymap(s) >> 21], 1u);
      }
      const float selk = (s > 0.0f && m > 0.5f) ? 1.0f : 0.0f;
#pragma unroll
      for (int k = 0; k < 5; ++k) {
        const float p = sigm(fget(kp4[k], j)) * selk;
        const float t = fget(kg4[k], j) * selk;
        ks[k * 3 + 0] = fmaf(p, t, ks[k * 3 + 0]);
        ks[k * 3 + 1] = fmaf(p, p, ks[k * 3 + 1]);
        ks[k * 3 + 2] = fmaf(t, t, ks[k * 3 + 2]);
      }
    }
  }

  // deterministic block partials for the 15 kernel-dice sums
  for (int v = 0; v < 15; ++v) {
    const float r = blockReduceF(ks[v], fbuf);
    if (threadIdx.x == 0)
      kpart[((size_t)pair * kNB + blockIdx.x) * 15 + v] = r;
  }
  // integer-atomic counts (order independent -> deterministic)
  const unsigned pr = blockReduceU(posc, ubuf);
  const unsigned nr = blockReduceU(negc, ubuf);
  if (threadIdx.x == 0) {
    atomicAdd(&ws_u[POS_OFF + pair], pr);
    atomicAdd(&ws_u[NEG_OFF + pair], nr);
  }
  __syncthreads();
  unsigned* gh = ws_u + HIST_OFF + (size_t)pair * BINS_A;
  for (int i = threadIdx.x; i < BINS_A; i += kT)
    if (hist[i]) atomicAdd(&gh[i], hist[i]);
}

// ---------------------------------------------------------------------------
// Radix-select step: scan this pair's histogram from the top, pick the bucket
// holding the r-th largest, update prefix/rank, then zero the histogram for
// the next refinement pass. One block per pair.
__global__ __launch_bounds__(kT) void k_select(unsigned* __restrict__ ws_u, int stage) {
  const int pair = blockIdx.x;
  unsigned* hist = ws_u + HIST_OFF + (size_t)pair * BINS_A;

  if (threadIdx.x == 0) {
    if (stage == 0) {
      const unsigned pos = ws_u[POS_OFF + pair];
      const unsigned neg = ws_u[NEG_OFF + pair];
      const unsigned nn  = pos * (unsigned)kRatio;
      const unsigned r   = nn < neg ? nn : neg;
      ws_u[FB_OFF + pair]   = (pos == 0u || r == 0u) ? 1u : 0u;
      ws_u[RANK_OFF + pair] = r;
      ws_u[PREF_OFF + pair] = 0u;
    }
    if (!ws_u[FB_OFF + pair]) {
      unsigned r = ws_u[RANK_OFF + pair];
      const int bins = (stage == 2) ? BINS_B : BINS_A;
      unsigned cum = 0; int bsel = 0;
      for (int i = bins - 1; i >= 0; --i) {
        const unsigned h = hist[i];
        if (cum + h >= r) { bsel = i; r -= cum; break; }
        cum += h;
      }
      unsigned pref = ws_u[PREF_OFF + pair];
      if (stage == 0) {
        pref = (unsigned)bsel;
      } else if (stage == 1) {
        pref = (pref << 11) | (unsigned)bsel;
      } else {
        const unsigned key = (pref << 10) | (unsigned)bsel;
        ((float*)ws_u)[THR_OFF + pair] = keyunmap(key);
      }
      ws_u[PREF_OFF + pair] = pref;
      ws_u[RANK_OFF + pair] = r;
    }
  }
  __syncthreads();
  for (int i = threadIdx.x; i < BINS_A; i += kT) hist[i] = 0u;
}

// ---------------------------------------------------------------------------
// Refinement histogram over the text channel only, restricted to keys whose
// high bits match the selected prefix. Chunk staged into LDS by the TDM.
__global__ __launch_bounds__(kT) void k_hist(const float* __restrict__ outp,
                                             const float* __restrict__ labp,
                                             unsigned* __restrict__ ws_u,
                                             int stage) {
  __shared__ unsigned hist[BINS_A];
#if HAVE_TDM
  __shared__ float4 stex[kF4PerBlock];
  __shared__ float4 sgt[kF4PerBlock];
#endif
  const int pair = blockIdx.y;
  if (ws_u[FB_OFF + pair]) return;       // uniform per block
  const int b = pair >> 1, c = pair & 1;
  const unsigned pref = ws_u[PREF_OFF + pair];
  const int bins = (stage == 2) ? BINS_B : BINS_A;

  for (int i = threadIdx.x; i < bins; i += kT) hist[i] = 0u;

  const int txtCh = c * kMaps + (kMaps - 1);
  const float4* texts = (const float4*)(outp + (size_t)(b * kC + txtCh) * kHW);
  const float4* gtex  = (const float4*)(labp + (size_t)(b * kC + txtCh) * kHW);
  const int base = blockIdx.x * kF4PerBlock;

#if HAVE_TDM
  if (threadIdx.x == 0) {   // one wave issues the DMAs; TDM ignores EXEC
    tdm_load_f32_1d((const float*)(texts + base), lds_offset_of(stex),
                    kF4PerBlock * 4);
    tdm_load_f32_1d((const float*)(gtex + base), lds_offset_of(sgt),
                    kF4PerBlock * 4);
    __builtin_amdgcn_s_wait_tensorcnt(0);
  }
  __syncthreads();
  const float4* S = stex;
  const float4* G = sgt;
#else
  __syncthreads();
  const float4* S = texts + base;
  const float4* G = gtex + base;
#endif

  for (int it = 0; it < kF4PerThread; ++it) {
    const int li = it * kT + threadIdx.x;
    const float4 s4 = S[li];
    const float4 g4 = G[li];
#pragma unroll
    for (int j = 0; j < 4; ++j) {
      if (fget(g4, j) <= 0.5f) {
        const unsigned k = keymap(fget(s4, j));
        if (stage == 1) {
          if ((k >> 21) == pref) atomicAdd(&hist[(k >> 10) & 2047u], 1u);
        } else {
          if ((k >> 10) == pref) atomicAdd(&hist[k & 1023u], 1u);
        }
      }
    }
  }
  __syncthreads();
  unsigned* gh = ws_u + HIST_OFF + (size_t)pair * BINS_A;
  for (int i = threadIdx.x; i < bins; i += kT)
    if (hist[i]) atomicAdd(&gh[i], hist[i]);
}

// ---------------------------------------------------------------------------
// Pass 2: text dice partial sums with the OHEM-selected mask. TDM-staged.
__global__ __launch_bounds__(kT) void k_textdice(const float* __restrict__ outp,
                                                 const float* __restrict__ labp,
                                                 const float* __restrict__ tmp,
                                                 const unsigned* __restrict__ ws_u,
                                                 float* __restrict__ tpart) {
  __shared__ float fbuf[kT];
#if HAVE_TDM
  __shared__ float4 stex[kF4PerBlock];
  __shared__ float4 sgt[kF4PerBlock];
  __shared__ float4 stm[kF4PerBlock];
#endif
  const int pair = blockIdx.y;
  const int b = pair >> 1, c = pair & 1;
  const unsigned fb = ws_u[FB_OFF + pair];
  const float thr = ((const float*)ws_u)[THR_OFF + pair];

  const int txtCh = c * kMaps + (kMaps - 1);
  const float4* texts = (const float4*)(outp + (size_t)(b * kC + txtCh) * kHW);
  const float4* gtex  = (const float4*)(labp + (size_t)(b * kC + txtCh) * kHW);
  const float4* tmv   = (const float4*)(tmp  + (size_t)b * kHW);
  const int base = blockIdx.x * kF4PerBlock;

#if HAVE_TDM
  if (threadIdx.x == 0) {
    tdm_load_f32_1d((const float*)(texts + base), lds_offset_of(stex),
                    kF4PerBlock * 4);
    tdm_load_f32_1d((const float*)(gtex + base), lds_offset_of(sgt),
                    kF4PerBlock * 4);
    tdm_load_f32_1d((const float*)(tmv + base), lds_offset_of(stm),
                    kF4PerBlock * 4);
    __builtin_amdgcn_s_wait_tensorcnt(0);
  }
  __syncthreads();
  const float4* S = stex;
  const float4* G = sgt;
  const float4* M = stm;
#else
  const float4* S = texts + base;
  const float4* G = gtex + base;
  const float4* M = tmv + base;
#endif

  float sa = 0.0f, sb = 0.0f, sc = 0.0f;
  for (int it = 0; it < kF4PerThread; ++it) {
    const int li = it * kT + threadIdx.x;
    const float4 s4 = S[li];
    const float4 g4 = G[li];
    const float4 m4 = M[li];
#pragma unroll
    for (int j = 0; j < 4; ++j) {
      const float s = fget(s4, j);
      const float g = fget(g4, j);
      const float m = fget(m4, j);
      float sel;
      if (fb) sel = m;  // fallback: raw (continuous) training mask
      else    sel = ((s >= thr || g > 0.5f) && m > 0.5f) ? 1.0f : 0.0f;
      const float p = sigm(s) * sel;
      const float t = g * sel;
      sa = fmaf(p, t, sa);
      sb = fmaf(p, p, sb);
      sc = fmaf(t, t, sc);
    }
  }
  const float ra = blockReduceF(sa, fbuf);
  const float rb = blockReduceF(sb, fbuf);
  const float rc = blockReduceF(sc, fbuf);
  if (threadIdx.x == 0) {
    float* dst = tpart + ((size_t)pair * kNB + blockIdx.x) * 3;
    dst[0] = ra; dst[1] = rb; dst[2] = rc;
  }
}

// ---------------------------------------------------------------------------
// Finalize. The 288 partial-sum reductions over kNB=100 blocks form
// S = R x ones — computed with chained full-f32 WMMAs (16 combos per chain,
// 4 blocks folded per instruction, accumulated in the f32 C operand).
__global__ __launch_bounds__(kT) void k_final(const float* __restrict__ kpart,
                                              const float* __restrict__ tpart,
                                              float* __restrict__ outv) {
  __shared__ float ksum[kPairs * 15];
  __shared__ float tsum[kPairs * 3];
  const int tid = threadIdx.x;
#if HAVE_WMMA_F32
  const int wave = tid >> 5;
  const int lane = tid & 31;
  const int half = lane >> 4;   // A-matrix K-halves live in lane halves
  const int row  = lane & 15;   // A-matrix row M
  const wm_v2f ones = {1.0f, 1.0f};
  // kpart: 240 sums = 15 chains of 16 rows; R[(pair,v)][blk] @ pair*1500+blk*15+v
  for (int ch = wave; ch < 15; ch += 8) {
    const int co = ch * 16 + row;
    const int pair = co / 15, v = co % 15;
    const float* base = kpart + (size_t)pair * kNB * 15 + v;
    wm_v8f acc = {};
    for (int i = 0; i < kNB / 4; ++i) {
      const int blk = i * 4 + half * 2;
      wm_v2f a;
      a.x = base[(size_t)(blk + 0) * 15];
      a.y = base[(size_t)(blk + 1) * 15];
      acc = __builtin_amdgcn_wmma_f32_16x16x4_f32(false, a, false, ones,
                                                  (short)0, acc, false, false);
    }
    if (row == 0) {   // D[m][*] replicated: lane0 has M=0..7, lane16 M=8..15
#pragma unroll
      for (int j = 0; j < 8; ++j) ksum[ch * 16 + half * 8 + j] = acc[j];
    }
  }
  // tpart: 48 sums = 3 chains of 16; R[(pair,v)][blk] @ pair*300+blk*3+v
  for (int ch = wave; ch < 3; ch += 8) {
    const int co = ch * 16 + row;
    const int pair = co / 3, v = co % 3;
    const float* base = tpart + (size_t)pair * kNB * 3 + v;
    wm_v8f acc = {};
    for (int i = 0; i < kNB / 4; ++i) {
      const int blk = i * 4 + half * 2;
      wm_v2f a;
      a.x = base[(size_t)(blk + 0) * 3];
      a.y = base[(size_t)(blk + 1) * 3];
      acc = __builtin_amdgcn_wmma_f32_16x16x4_f32(false, a, false, ones,
                                                  (short)0, acc, false, false);
    }
    if (row == 0) {
#pragma unroll
      for (int j = 0; j < 8; ++j) tsum[ch * 16 + half * 8 + j] = acc[j];
    }
  }
#else
  if (tid < kPairs * 15) {
    float s = 0.0f;
    const int pair = tid / 15, v = tid % 15;
    for (int blk = 0; blk < kNB; ++blk)
      s += kpart[((size_t)pair * kNB + blk) * 15 + v];
    ksum[tid] = s;
  }
  if (tid < kPairs * 3) {
    float s = 0.0f;
    const int pair = tid / 3, v = tid % 3;
    for (int blk = 0; blk < kNB; ++blk)
      s += tpart[((size_t)pair * kNB + blk) * 3 + v];
    tsum[tid] = s;
  }
#endif
  __syncthreads();
  if (tid == 0) {
    float loss_sum = 0.0f, lt_sum = 0.0f, lk_sum = 0.0f;
    for (int c = 0; c < 2; ++c) {
      float lt = 0.0f, lk = 0.0f;
      for (int b = 0; b < kB; ++b) {
        const int pair = b * 2 + c;
        const float ta = tsum[pair * 3 + 0];
        const float tb = tsum[pair * 3 + 1];
        const float tc = tsum[pair * 3 + 2];
        lt += 1.0f - 2.0f * ta / ((tb + kEps) + (tc + kEps));
        float lkb = 0.0f;
        for (int k = 0; k < 5; ++k) {
          const float ka = ksum[pair * 15 + k * 3 + 0];
          const float kb = ksum[pair * 15 + k * 3 + 1];
          const float kc = ksum[pair * 15 + k * 3 + 2];
          lkb += 1.0f - 2.0f * ka / ((kb + kEps) + (kc + kEps));
        }
        lk += lkb / 5.0f;
      }
      lt /= (float)kB;
      lk /= (float)kB;
      loss_sum += kLambda * lt + (1.0f - kLambda) * lk;
      lt_sum += lt;
      lk_sum += lk;
    }
    outv[0] = loss_sum * 0.5f;
    outv[1] = lt_sum * 0.5f;
    outv[2] = lk_sum * 0.5f;
  }
}

}  // namespace

extern "C" void kernel_launch(void* const* d_in, const int* in_sizes, int n_in,
                              void* d_out, int out_size, void* d_ws, size_t ws_size,
                              hipStream_t stream) {
  (void)in_sizes; (void)n_in; (void)out_size; (void)ws_size;
  const float* outp = (const float*)d_in[0];   // outputs (8,12,640,640) f32
  const float* labp = (const float*)d_in[1];   // labels  (8,12,640,640) f32
  const float* tmp_ = (const float*)d_in[2];   // training_mask (8,1,640,640) f32
  unsigned* ws_u = (unsigned*)d_ws;
  float*    ws_f = (float*)d_ws;
  float* kpart = ws_f + KPART_OFF;
  float* tpart = ws_f + TPART_OFF;

  const dim3 gridS(kNB, kPairs);

  k_zero<<<(ZERO_WORDS + kT - 1) / kT, kT, 0, stream>>>(ws_u, ZERO_WORDS);
  k_stats<<<gridS, kT, 0, stream>>>(outp, labp, tmp_, ws_u, kpart);
  k_select<<<kPairs, kT, 0, stream>>>(ws_u, 0);
  k_hist<<<gridS, kT, 0, stream>>>(outp, labp, ws_u, 1);
  k_select<<<kPairs, kT, 0, stream>>>(ws_u, 1);
  k_hist<<<gridS, kT, 0, stream>>>(outp, labp, ws_u, 2);
  k_select<<<kPairs, kT, 0, stream>>>(ws_u, 2);
  k_textdice<<<gridS, kT, 0, stream>>>(outp, labp, tmp_, ws_u, tpart);
  k_final<<<1, kT, 0, stream>>>(kpart, tpart, (float*)d_out);
}